// ApsDown_420906795251
// MI455X (gfx1250) — compile-verified
//
#include <hip/hip_runtime.h>
#include <hip/hip_bf16.h>
#include <stdint.h>

// Problem constants (from reference): x[16,64,256,256] f32
#define B_   16
#define C_   64
#define N_   256
#define HO_  128
#define OUT_ELEMS (B_ * C_ * HO_ * HO_)   // 16,777,216 floats, then 16 idx floats

// ---------------------------------------------------------------------------
// Async global->LDS tile loader (CDNA5 GLOBAL_LOAD_ASYNC_TO_LDS_B128 path).
// Loads 34 rows x 256 cols of one (b,c) plane, rows (r0-1 .. r0+32) with
// circular wrap (&255), into tile_s laid out [row*256 + col].
// Streaming data (268MB/pass vs 192MB L2, no reuse) -> non-temporal hint.
// Per-lane: LDS dest byte addr in %0, 32-bit global byte offset in %1,
// uniform 64-bit base in SGPR pair %2.  Completion via s_wait_asynccnt.
// ---------------------------------------------------------------------------
__device__ __forceinline__ void async_load_tile(const float* __restrict__ plane,
                                                int r0, float* tile_s, int tid) {
    // Low 32 bits of a generic pointer to LDS are the LDS byte address
    // (ISA 10.2 aperture).
    uint32_t lds_base = (uint32_t)(uintptr_t)tile_s;
    // 34 rows * 64 float4-segments = 2176 lane-chunks of 16B
    for (int q = tid; q < 34 * 64; q += 256) {
        int row = q >> 6;
        int seg = q & 63;
        uint32_t grow = (uint32_t)((r0 - 1 + row) & (N_ - 1));
        uint32_t goff = grow * (N_ * 4u) + (uint32_t)seg * 16u;
        uint32_t loff = lds_base + (uint32_t)row * (N_ * 4u) + (uint32_t)seg * 16u;
        asm volatile("global_load_async_to_lds_b128 %0, %1, %2 th:TH_LOAD_NT"
                     :
                     : "v"(loff), "v"(goff), "s"(plane)
                     : "memory");
    }
    asm volatile("s_wait_asynccnt 0" ::: "memory");
    __syncthreads();
}

// horizontal 3-tap, scalar form (circular columns via caller's wl/w/wr)
__device__ __forceinline__ float hrow3(const float* ts, int r, int wl, int w, int wr) {
    return ts[r * N_ + wl] + 2.0f * ts[r * N_ + w] + ts[r * N_ + wr];
}

// ---------------------------------------------------------------------------
// Kernel 0: zero the 16x4 phase-norm accumulators in workspace
// ---------------------------------------------------------------------------
__global__ void aps_zero_norms(float* __restrict__ norms) {
    int t = threadIdx.x;
    if (t < B_ * 4) norms[t] = 0.0f;
}

// ---------------------------------------------------------------------------
// Kernel 1: blur + per-(batch,phase) sum of squares.
// grid = B*C*8 blocks of 256 threads; block = one (b,c) plane, 32-row band.
// Thread layout: cg = tid&63 owns 4 adjacent columns (one ds_load_b128 + two
// edge b32 per LDS row -> 6B LDS traffic/pixel), rsub = tid>>6 owns 8 rows
// with a vertical sliding window on the horizontal partial sums.
// phase = (h&1) + 2*(w&1)  (matches reference stack order ee,oe,eo,oo).
// ---------------------------------------------------------------------------
__global__ void __launch_bounds__(256)
aps_norms_kernel(const float* __restrict__ x, float* __restrict__ norms) {
    __shared__ __align__(16) float tile_s[34 * N_];
    __shared__ float red[4];

    int blk  = blockIdx.x;
    int tile = blk & 7;            // 8 row-bands of 32
    int c    = (blk >> 3) & (C_ - 1);
    int b    = blk >> 9;           // /(64*8)
    int r0   = tile * 32;

    const float* plane = x + (((size_t)b * C_ + c) * N_ * N_);
    int tid = threadIdx.x;

    async_load_tile(plane, r0, tile_s, tid);
    if (tid < 4) red[tid] = 0.0f;

    int cg   = tid & 63;           // column group: cols 4cg .. 4cg+3 (c0 even)
    int rsub = tid >> 6;           // 0..3 -> band-rows 8rsub .. 8rsub+7
    int c0 = cg * 4;
    int cl = (c0 - 1) & (N_ - 1);
    int cr = (c0 + 4) & (N_ - 1);

    // horizontal partial sums for 4 columns of LDS row r
    auto hrow4 = [&](int r) -> float4 {
        const float4 v = *(const float4*)&tile_s[r * N_ + c0];
        float l  = tile_s[r * N_ + cl];
        float rr = tile_s[r * N_ + cr];
        float4 h;
        h.x = l   + 2.0f * v.x + v.y;
        h.y = v.x + 2.0f * v.y + v.z;
        h.z = v.y + 2.0f * v.z + v.w;
        h.w = v.z + 2.0f * v.w + rr;
        return h;
    };

    int rbase = 8 * rsub;                      // even -> row parity == k&1
    float4 hp = hrow4(rbase + 0);              // window top    (band-row k-1)
    float4 hc = hrow4(rbase + 1);              // window center (band-row k)
    float a0 = 0.0f, a1 = 0.0f, a2 = 0.0f, a3 = 0.0f;  // phases 0..3
    #pragma unroll
    for (int k = 0; k < 8; ++k) {
        float4 hn = hrow4(rbase + k + 2);
        float yx = (hp.x + 2.0f * hc.x + hn.x) * 0.0625f;
        float yy = (hp.y + 2.0f * hc.y + hn.y) * 0.0625f;
        float yz = (hp.z + 2.0f * hc.z + hn.z) * 0.0625f;
        float yw = (hp.w + 2.0f * hc.w + hn.w) * 0.0625f;
        float se = yx * yx + yz * yz;          // even columns -> phases 0/1
        float so = yy * yy + yw * yw;          // odd  columns -> phases 2/3
        if (k & 1) { a1 += se; a3 += so; }     // odd absolute row
        else       { a0 += se; a2 += so; }     // even absolute row
        hp = hc;
        hc = hn;
    }

    __syncthreads();
    atomicAdd(&red[0], a0);
    atomicAdd(&red[1], a1);
    atomicAdd(&red[2], a2);
    atomicAdd(&red[3], a3);
    __syncthreads();
    if (tid < 4) atomicAdd(&norms[b * 4 + tid], red[tid]);
}

// ---------------------------------------------------------------------------
// Kernel 2: per-batch argmax over 4 phases (first-max tie-break, jnp.argmax).
// Writes int idx to ws and float idx to d_out tail.
// ---------------------------------------------------------------------------
__global__ void aps_argmax_kernel(const float* __restrict__ norms,
                                  int* __restrict__ idx_ws,
                                  float* __restrict__ idx_out) {
    int b = threadIdx.x;
    if (b < B_) {
        float best = norms[b * 4];
        int   bi   = 0;
        for (int p = 1; p < 4; ++p) {
            float v = norms[b * 4 + p];
            if (v > best) { best = v; bi = p; }
        }
        idx_ws[b]  = bi;
        idx_out[b] = (float)bi;
    }
}

// ---------------------------------------------------------------------------
// Kernel 3: recompute blur at the winning phase's sites and gather to d_out.
// grid = B*C*8 blocks of 256; block handles 16 output rows (32 input rows+halo).
// out[b,c,i,j] = blurred[b,c, 2i+dr, 2j+dc]  with phase=idx[b], dr=phase&1,
// dc=(phase>>1)&1.  LDS traffic here is already below the HBM floor.
// ---------------------------------------------------------------------------
__global__ void __launch_bounds__(256)
aps_gather_kernel(const float* __restrict__ x, const int* __restrict__ idx_ws,
                  float* __restrict__ out) {
    __shared__ __align__(16) float tile_s[34 * N_];

    int blk  = blockIdx.x;
    int ot   = blk & 7;            // output row-tile of 16
    int c    = (blk >> 3) & (C_ - 1);
    int b    = blk >> 9;
    int i0   = ot * 16;
    int r0   = 2 * i0;             // tile rows cover input r0-1 .. r0+32

    const float* plane = x + (((size_t)b * C_ + c) * N_ * N_);
    int tid = threadIdx.x;

    async_load_tile(plane, r0, tile_s, tid);

    int phase = idx_ws[b];
    int dr = phase & 1;
    int dc = (phase >> 1) & 1;

    float* oplane = out + (((size_t)b * C_ + c) * HO_ * HO_);

    #pragma unroll
    for (int k = 0; k < 8; ++k) {
        int p  = tid + 256 * k;          // 0 .. 2047
        int il = p >> 7;                 // 0 .. 15
        int j  = p & (HO_ - 1);          // 0 .. 127
        int rc = 2 * il + dr + 1;        // LDS row of stencil center (row0 = r0-1)
        int w  = 2 * j + dc;
        int wl = (w - 1) & (N_ - 1);
        int wr = (w + 1) & (N_ - 1);
        float y = (hrow3(tile_s, rc - 1, wl, w, wr)
                   + 2.0f * hrow3(tile_s, rc, wl, w, wr)
                   + hrow3(tile_s, rc + 1, wl, w, wr)) * 0.0625f;
        __builtin_nontemporal_store(y, &oplane[(size_t)(i0 + il) * HO_ + j]);
    }
}

// ---------------------------------------------------------------------------
extern "C" void kernel_launch(void* const* d_in, const int* in_sizes, int n_in,
                              void* d_out, int out_size, void* d_ws, size_t ws_size,
                              hipStream_t stream) {
    (void)in_sizes; (void)n_in; (void)out_size; (void)ws_size;

    const float* x = (const float*)d_in[0];
    // d_in[1] is the 3x3 binomial filter; it is exactly [1,2,1]⊗[1,2,1]/16,
    // which the kernels implement separably in registers.

    float* out     = (float*)d_out;
    float* idx_out = out + OUT_ELEMS;          // 16 idx values as floats

    float* norms  = (float*)d_ws;              // 64 floats
    int*   idx_ws = (int*)d_ws + 64;           // 16 ints

    aps_zero_norms<<<1, 64, 0, stream>>>(norms);

    dim3 grid(B_ * C_ * 8);                    // 8192
    dim3 block(256);
    aps_norms_kernel<<<grid, block, 0, stream>>>(x, norms);

    aps_argmax_kernel<<<1, 32, 0, stream>>>(norms, idx_ws, idx_out);

    aps_gather_kernel<<<grid, block, 0, stream>>>(x, idx_ws, out);
}